// SpecRadLoss_32555852103918
// MI455X (gfx1250) — compile-verified
//
#include <hip/hip_runtime.h>
#include <math.h>

typedef __attribute__((ext_vector_type(2))) float v2f;
typedef __attribute__((ext_vector_type(8))) float v8f;

#define BATCH 8
#define LEN   2048
#define SCOL  128
#define KC    32
#define LDP   34   // LDS row stride (floats): even -> 8B-aligned v2f reads, padded vs 32 banks

// ---------------------------------------------------------------------------
// sp = sphere / ||sphere||_col ; y0[b] = sp  (broadcast)
// ---------------------------------------------------------------------------
__global__ __launch_bounds__(128) void normalize_bcast(const float* __restrict__ sphere,
                                                       float* __restrict__ sp,
                                                       float* __restrict__ y0) {
    const int s = blockIdx.x;      // column 0..127
    const int t = threadIdx.x;     // 0..127
    __shared__ float red[128];
    float sum = 0.f;
    for (int l = t; l < LEN; l += 128) {
        float v = sphere[(size_t)l * SCOL + s];
        sum += v * v;
    }
    red[t] = sum;
    __syncthreads();
    for (int off = 64; off > 0; off >>= 1) {
        if (t < off) red[t] += red[t + off];
        __syncthreads();
    }
    const float inv = 1.0f / sqrtf(red[0]);
    for (int l = t; l < LEN; l += 128) {
        float v = sphere[(size_t)l * SCOL + s] * inv;
        sp[(size_t)l * SCOL + s] = v;
        #pragma unroll
        for (int b = 0; b < BATCH; ++b)
            y0[((size_t)b * LEN + l) * SCOL + s] = v;
    }
}

// ---------------------------------------------------------------------------
// Generated-matrix GEMM:
//   MODE 0: A[i][j] = vec[b][(i-j) mod L]   (circulant, == FFT circular conv)
//   MODE 1: A[i][j] = vec[b][|i-j|]         (symmetric Toeplitz)
//   SUB 0 : Out = A @ In
//   SUB 1 : Out = Ysub - A @ In
// Block: 128 threads = 4 waves. Block tile: 64 rows x 128 cols.
// Wave tile: 16 rows x 128 cols = 8 x (16x16) WMMA_F32_16X16X4_F32 accumulators.
// ---------------------------------------------------------------------------
template <int MODE, bool SUB>
__global__ __launch_bounds__(128) void gemm_gen(const float* __restrict__ vec,
                                                const float* __restrict__ In,
                                                const float* __restrict__ Ysub,
                                                float* __restrict__ Out) {
    __shared__ float lds[SCOL * LDP];          // y-panel, transposed: lds[col*LDP + krow]

    const int b    = blockIdx.y;
    const int tid  = threadIdx.x;
    const int wave = tid >> 5;
    const int lane = tid & 31;
    const int half = lane >> 4;                // 0: lanes 0-15, 1: lanes 16-31
    const int r    = lane & 15;
    const int m0   = blockIdx.x * 64 + wave * 16;
    const int i    = m0 + r;                   // A-matrix row held by this lane

    const float* vb  = vec + (size_t)b * LEN;                // 8 KB generator vector
    const float* Inb = In  + (size_t)b * LEN * SCOL;

    const v8f vzero = {0.f, 0.f, 0.f, 0.f, 0.f, 0.f, 0.f, 0.f};
    v8f acc[8];
    #pragma unroll
    for (int nt = 0; nt < 8; ++nt) acc[nt] = vzero;

    for (int k = 0; k < LEN; k += KC) {
        __syncthreads();
        // stage In[k .. k+31][0..127] transposed into LDS (thread == column)
        #pragma unroll
        for (int kr = 0; kr < KC; ++kr)
            lds[tid * LDP + kr] = Inb[(size_t)(k + kr) * SCOL + tid];
        if (k + KC < LEN) {                     // prefetch next chunk (global_prefetch_b8)
            #pragma unroll
            for (int kr = 0; kr < KC; kr += 8)
                __builtin_prefetch(&Inb[(size_t)(k + KC + kr) * SCOL + tid], 0, 1);
        }
        __syncthreads();

        #pragma unroll
        for (int kk = 0; kk < KC; kk += 4) {
            // A fragment (16x4 f32): lane holds A[i][j], A[i][j+1]; K = vgpr + 2*half
            const int j = k + kk + 2 * half;
            float ax, ay;
            if (MODE == 0) {
                ax = vb[(i - j)     & (LEN - 1)];
                ay = vb[(i - j - 1) & (LEN - 1)];
            } else {
                int d0 = i - j;
                int d1 = d0 - 1;
                ax = vb[d0 < 0 ? -d0 : d0];
                ay = vb[d1 < 0 ? -d1 : d1];
            }
            v2f a;
            a.x = ax;
            a.y = ay;

            const int bofs = kk + 2 * half;    // B fragment rows: K = vgpr + 2*half
            #pragma unroll
            for (int nt = 0; nt < 8; ++nt) {
                // B (4x16 f32): lane r holds B[bofs][n0+r], B[bofs+1][n0+r]
                const v2f bf = *(const v2f*)&lds[(nt * 16 + r) * LDP + bofs];
                acc[nt] = __builtin_amdgcn_wmma_f32_16x16x4_f32(
                    /*neg_a=*/false, a, /*neg_b=*/false, bf,
                    /*c_mod=*/(short)0, acc[nt],
                    /*reuse_a=*/false, /*reuse_b=*/false);
            }
        }
    }

    // D layout (16x16 f32): vgpr v, lane -> row m0 + v + 8*half, col n0 + r
    const size_t base = (size_t)b * LEN * SCOL;
    #pragma unroll
    for (int nt = 0; nt < 8; ++nt) {
        #pragma unroll
        for (int v = 0; v < 8; ++v) {
            const int row = m0 + v + 8 * half;
            const int col = nt * 16 + r;
            const size_t off = base + (size_t)row * SCOL + col;
            float val = acc[nt][v];
            if (SUB) val = Ysub[off] - val;
            Out[off] = val;
        }
    }
}

// ---------------------------------------------------------------------------
// out = mean_b max_s ( ||y[b,:,s] + 1e-12*sp[:,s]||_2 )^(1/10)
// f32 sum-of-squares on purpose: matches JAX f32 norm (incl. overflow behavior)
// ---------------------------------------------------------------------------
__global__ __launch_bounds__(128) void finalize_k(const float* __restrict__ y,
                                                  const float* __restrict__ sp,
                                                  float* __restrict__ out) {
    const int b = blockIdx.x;
    const int s = threadIdx.x;                 // one thread per column
    float ss = 0.f;
    for (int l = 0; l < LEN; ++l) {
        float v = y[((size_t)b * LEN + l) * SCOL + s] + 1e-12f * sp[(size_t)l * SCOL + s];
        ss += v * v;
    }
    float val = powf(ss, 0.05f);               // (sqrt(ss))^(1/10)
    __shared__ float red[128];
    red[s] = val;
    __syncthreads();
    for (int off = 64; off > 0; off >>= 1) {
        if (s < off) red[s] = fmaxf(red[s], red[s + off]);
        __syncthreads();
    }
    if (s == 0) atomicAdd(out, red[0] * (1.0f / BATCH));
}

__global__ void zero_out_k(float* out) { out[0] = 0.f; }

// ---------------------------------------------------------------------------
extern "C" void kernel_launch(void* const* d_in, const int* in_sizes, int n_in,
                              void* d_out, int out_size, void* d_ws, size_t ws_size,
                              hipStream_t stream) {
    const float* t_vec  = (const float*)d_in[0];   // (B, L)
    const float* c_inv  = (const float*)d_in[1];   // (B, L)
    const float* sphere = (const float*)d_in[2];   // (L, S)
    float* out = (float*)d_out;

    float* ws = (float*)d_ws;
    float* sp = ws;                                   // L*S
    float* yA = sp + (size_t)LEN * SCOL;              // B*L*S
    float* yB = yA + (size_t)BATCH * LEN * SCOL;      // B*L*S
    float* cy = yB + (size_t)BATCH * LEN * SCOL;      // B*L*S   (~25 MB total)

    normalize_bcast<<<dim3(SCOL), dim3(128), 0, stream>>>(sphere, sp, yA);

    const dim3 ggrid(LEN / 64, BATCH);
    const float* cur = yA;
    for (int it = 0; it < 10; ++it) {
        float* nxt = (it & 1) ? yA : yB;
        // cy = C @ y          (circulant == FFT circular convolution)
        gemm_gen<0, false><<<ggrid, dim3(128), 0, stream>>>(c_inv, cur, nullptr, cy);
        // y' = y - T @ cy     (Toeplitz), fused subtract
        gemm_gen<1, true ><<<ggrid, dim3(128), 0, stream>>>(t_vec, cy, cur, nxt);
        cur = nxt;
    }
    // after 10 iterations cur == yA

    zero_out_k<<<1, 1, 0, stream>>>(out);
    finalize_k<<<dim3(BATCH), dim3(128), 0, stream>>>(cur, sp, out);
}